// _HybridQuantumHead_65481071403395
// MI455X (gfx1250) — compile-verified
//
#include <hip/hip_runtime.h>
#include <hip/hip_bf16.h>

// HybridQuantumHead for MI455X (gfx1250).
//
// Closed form: the variational layers are token-independent, so
//   out[t,q] = cos(x[t,q])*A_q - sin(x[t,q])*B_q,  A_q=<s|Z_q|s>, B_q=<s|X_q|s>
// Stage 1: simulate the 10-qubit base circuit once (LDS), reduce A/B, build
//          W (32x16) block-diagonal in d_ws.
// Stage 2: Phi(4096x32) @ W(32x16) via V_WMMA_F32_16X16X4_F32, one wave per
//          16x16 output tile, 8 K-chunks of 4. sin/cos via hardware
//          transcendentals (x ~ N(0,1), well inside the fast range).

#define NQ 10
#define NL 4
#define DIM 1024  // 2^NQ

typedef __attribute__((ext_vector_type(2))) float v2f;
typedef __attribute__((ext_vector_type(8))) float v8f;

// ---------------- Stage 1: base statevector + A/B -> W -------------------
__global__ __launch_bounds__(1024) void qh_base_state(
    const float* __restrict__ params, float* __restrict__ W) {
  __shared__ float st[DIM];
  __shared__ float partA[32 * NQ];
  __shared__ float partB[32 * NQ];
  __shared__ float ab[2 * NQ];

  const int k = threadIdx.x;
  st[k] = (k == 0) ? 1.0f : 0.0f;
  __syncthreads();

  // qubit q <-> bit (NQ-1-q) of the flat index (wire 0 most significant)
  for (int l = 0; l < NL; ++l) {
    // RY(params[l][q]) on every qubit
    for (int q = 0; q < NQ; ++q) {
      const float th = 0.5f * params[l * NQ + q];
      float s, c;
      sincosf(th, &s, &c);  // precise; one-shot kernel feeding all tokens
      const int mask = 1 << (NQ - 1 - q);
      const float s0 = st[k & ~mask];
      const float s1 = st[k | mask];
      __syncthreads();
      st[k] = ((k & mask) == 0) ? (c * s0 - s * s1) : (s * s0 + c * s1);
      __syncthreads();
    }
    // CNOT ring: (0,1)...(8,9),(9,0)
    for (int g = 0; g < NQ; ++g) {
      const int ctrl = g;
      const int tgt = (g + 1) % NQ;
      const int cmask = 1 << (NQ - 1 - ctrl);
      const int tmask = 1 << (NQ - 1 - tgt);
      const int src = (k & cmask) ? (k ^ tmask) : k;
      const float v = st[src];
      __syncthreads();
      st[k] = v;
      __syncthreads();
    }
  }

  // Per-thread contributions: A_q = sum sign_q(k)*s_k^2 ; B_q = sum s_k*s_{k^mask}
  const float sk = st[k];
  const float p = sk * sk;
  float av[NQ], bv[NQ];
#pragma unroll
  for (int q = 0; q < NQ; ++q) {
    const int mask = 1 << (NQ - 1 - q);
    av[q] = (k & mask) ? -p : p;
    bv[q] = sk * st[k ^ mask];
  }

  const int wave = k >> 5;
  const int lane = k & 31;
#pragma unroll
  for (int q = 0; q < NQ; ++q) {
    float a = av[q], b = bv[q];
#pragma unroll
    for (int off = 16; off > 0; off >>= 1) {
      a += __shfl_down(a, off, 32);
      b += __shfl_down(b, off, 32);
    }
    if (lane == 0) {
      partA[wave * NQ + q] = a;
      partB[wave * NQ + q] = b;
    }
  }
  __syncthreads();

  // Deterministic final reduction across the 32 waves
  if (k < 2 * NQ) {
    const int q = k % NQ;
    const bool isB = (k >= NQ);
    float sum = 0.0f;
    for (int w = 0; w < 32; ++w)
      sum += isB ? partB[w * NQ + q] : partA[w * NQ + q];
    ab[k] = sum;
  }
  __syncthreads();

  // W (32x16 row-major): W[2q][q] = A_q, W[2q+1][q] = -B_q, else 0
  if (k < 32 * 16) {
    const int r = k >> 4;
    const int c = k & 15;
    float val = 0.0f;
    if (c < NQ) {
      if (r == 2 * c) val = ab[c];
      else if (r == 2 * c + 1) val = -ab[NQ + c];
    }
    W[r * 16 + c] = val;
  }
}

// ---------------- Stage 2: Phi @ W via WMMA f32 16x16x4 ------------------
// One wave per 16x16 output tile. K = 32 -> 8 chunks of 4.
// A (16x4):  lane L: m=L%16, h=L/16, a[v] = Phi[row][4c + 2h + v]
// B (4x16):  lane L: n=L%16,          b[v] = W[4c + 2h + v][n]
// D (16x16): lane L: n=L%16,          d[v] = D[v + 8h][n]
__global__ __launch_bounds__(128) void qh_encode_gemm(
    const float* __restrict__ x, const float* __restrict__ W,
    float* __restrict__ out) {
  const int lane = threadIdx.x & 31;
  const int wid = threadIdx.x >> 5;
  const int tile = blockIdx.x * 4 + wid;  // 0..255 (4096 rows / 16)
  const int m = lane & 15;
  const int h = lane >> 4;
  const int row = tile * 16 + m;

  v8f acc = {};
#pragma unroll
  for (int c = 0; c < 8; ++c) {
    // Phi[row][2q] = cos(x[row][q]), Phi[row][2q+1] = sin(x[row][q]); q = 2c+h
    const int q = 2 * c + h;
    const float xv = x[row * 64 + q];
    float sv, cv;
    __sincosf(xv, &sv, &cv);  // v_sin_f32/v_cos_f32; |x| small -> accurate
    v2f a;
    a.x = cv;
    a.y = sv;
    const int r0 = 4 * c + 2 * h;
    v2f b;
    b.x = W[r0 * 16 + m];
    b.y = W[(r0 + 1) * 16 + m];
    acc = __builtin_amdgcn_wmma_f32_16x16x4_f32(
        false, a, false, b, (short)0, acc, false, false);
  }

#pragma unroll
  for (int v = 0; v < 8; ++v) {
    const int orow = tile * 16 + v + 8 * h;
    out[orow * 64 + m] = acc[v];
  }
}

extern "C" void kernel_launch(void* const* d_in, const int* in_sizes, int n_in,
                              void* d_out, int out_size, void* d_ws,
                              size_t ws_size, hipStream_t stream) {
  (void)in_sizes; (void)n_in; (void)ws_size;
  const float* x = (const float*)d_in[0];       // (4,1024,64) f32
  const float* params = (const float*)d_in[1];  // (4,10) f32
  float* out = (float*)d_out;                   // (4,1024,64) f32
  float* W = (float*)d_ws;                      // 32x16 f32 scratch

  // Columns 16..63 are zero; memset everything, GEMM overwrites cols 0..15.
  (void)hipMemsetAsync(d_out, 0, (size_t)out_size * sizeof(float), stream);

  hipLaunchKernelGGL(qh_base_state, dim3(1), dim3(1024), 0, stream, params, W);
  hipLaunchKernelGGL(qh_encode_gemm, dim3(64), dim3(128), 0, stream, x, W, out);
}